// Self_fusegt_52501680227014
// MI455X (gfx1250) — compile-verified
//
#include <hip/hip_runtime.h>
#include <cstdint>

// Problem constants (B,C,H,W) = (8,64,256,256), fp32 in/out.
constexpr int kB = 8, kC = 64, kH = 256, kW = 256;
constexpr int kHW  = kH * kW;       // 65536
constexpr int kCHW = kC * kHW;

typedef unsigned int u32x4 __attribute__((ext_vector_type(4)));
typedef int          i32x8 __attribute__((ext_vector_type(8)));
typedef int          i32x4 __attribute__((ext_vector_type(4)));

#if defined(__HIP_DEVICE_COMPILE__) && defined(__gfx1250__)
  #if __has_builtin(__builtin_amdgcn_tensor_load_to_lds)
    #define USE_TDM 1
  #else
    #define USE_TDM 0
  #endif
  #if __has_builtin(__builtin_amdgcn_global_load_async_to_lds_b128)
    #define USE_ASYNC 1
  #else
    #define USE_ASYNC 0
  #endif
#else
  #define USE_TDM 0
  #define USE_ASYNC 0
#endif

#if USE_ASYNC
typedef __attribute__((address_space(1))) i32x4 as1_i32x4;   // global v4i32
typedef __attribute__((address_space(3))) i32x4 as3_i32x4;   // LDS v4i32
#endif

__device__ __forceinline__ void wait_tensorcnt0() {
#if defined(__HIP_DEVICE_COMPILE__) && defined(__gfx1250__)
  #if __has_builtin(__builtin_amdgcn_s_wait_tensorcnt)
    __builtin_amdgcn_s_wait_tensorcnt(0);
  #else
    asm volatile("s_wait_tensorcnt 0x0" ::: "memory");
  #endif
#endif
}

__device__ __forceinline__ void wait_asynccnt0() {
#if USE_ASYNC
  #if __has_builtin(__builtin_amdgcn_s_wait_asynccnt)
    __builtin_amdgcn_s_wait_asynccnt(0);
  #else
    asm volatile("s_wait_asynccnt 0x0" ::: "memory");
  #endif
#endif
}

// Stage three input rows (h-1,h,h+1) of channel c into in_lds[buf].
// Threads 0..191: one 16-byte chunk each (3 rows x 64 chunks). Rows outside
// [0,H) are zero-filled (zero padding in the reference).
__device__ __forceinline__ void stage_rows(const float* __restrict__ in,
                                           float (&in_lds)[2][3][kW],
                                           int b, int c, int h, int buf, int t) {
  if (t < 192) {
    const int r  = t >> 6;          // row 0..2
    const int i  = (t & 63) << 2;   // column 0..252, step 4
    const int hh = h - 1 + r;
    float* dst = &in_lds[buf][r][i];
    if ((unsigned)hh < (unsigned)kH) {
      const float* src = in + ((size_t)(b * kC + c) * kH + hh) * kW + i;
#if USE_ASYNC
      __builtin_amdgcn_global_load_async_to_lds_b128(
          (as1_i32x4*)(uintptr_t)src,
          (as3_i32x4*)(unsigned)(uintptr_t)dst,
          0, 0);
#else
      *(float4*)dst = *(const float4*)src;
#endif
    } else {
      *(float4*)dst = make_float4(0.f, 0.f, 0.f, 0.f);
    }
  }
}

__global__ __launch_bounds__(256) void fuse8_kernel(const float* __restrict__ in,
                                                    const float* __restrict__ gt,
                                                    float* __restrict__ out) {
  const int t  = threadIdx.x;       // w coordinate, 0..255
  const int bh = blockIdx.x;        // 0..B*H-1
  const int b  = bh >> 8;
  const int h  = bh & (kH - 1);

  __shared__ __align__(16) float gt_lds[8][kW];        // 8 KB, loaded once (TDM)
  __shared__ __align__(16) float in_lds[2][3][kW];     // 6 KB, double-buffered rows

#if USE_TDM
  // One TDM descriptor: 2D tile, 256 elems x 8 rows of fp32, row stride H*W elems,
  // pulls gt[b, 0:8, h, 0:256] into gt_lds in a single tensor_load_to_lds.
  if (t == 0) {
    const unsigned lds_off = (unsigned)(uintptr_t)&gt_lds[0][0];
    const unsigned long long ga =
        (unsigned long long)(uintptr_t)(gt + (size_t)b * 8 * kHW + (size_t)h * kW);
    u32x4 g0;
    g0[0] = 1u;                                            // count=1 (valid user D#)
    g0[1] = lds_off;                                       // lds_addr (bytes)
    g0[2] = (unsigned)(ga & 0xffffffffull);                // global_addr[31:0]
    g0[3] = (unsigned)((ga >> 32) & 0x01ffffffull)         // global_addr[56:32]
            | (2u << 30);                                  // type=2 ("image")
    i32x8 g1;
    g1[0] = (int)(2u << 16);                               // data_size=2 -> 4 bytes
    g1[1] = (int)((unsigned)(kW & 0xffff) << 16);          // tensor_dim0[15:0]
    g1[2] = (int)(((unsigned)kW >> 16) | (8u << 16));      // dim0[31:16] | dim1[15:0]=8
    g1[3] = (int)((unsigned)kW << 16);                     // dim1[31:16]=0 | tile_dim0=256
    g1[4] = 8;                                             // tile_dim1=8, tile_dim2=0
    g1[5] = kHW;                                           // tensor_dim0_stride[31:0]
    g1[6] = 0;                                             // stride0[47:32], stride1 lo
    g1[7] = 0;
    const i32x4 z4 = {0, 0, 0, 0};
    const i32x8 z8 = {0, 0, 0, 0, 0, 0, 0, 0};
    __builtin_amdgcn_tensor_load_to_lds(g0, g1, z4, z4, z8, 0);
  }
#else
  for (int d = 0; d < 8; ++d)
    gt_lds[d][t] = gt[((size_t)b * 8 + d) * kHW + (size_t)h * kW + t];
#endif

  // Kick off staging of channel 0 while the TDM transfer runs.
  stage_rows(in, in_lds, b, /*c=*/0, h, /*buf=*/0, t);

#if USE_TDM
  wait_tensorcnt0();     // only wave 0 has TENSORcnt>0; harmless elsewhere
#endif
  __syncthreads();

  // Per-pixel affinity weights, reused across all 64 channels.
  const float g0r = gt_lds[0][t], g1r = gt_lds[1][t], g2r = gt_lds[2][t],
              g3r = gt_lds[3][t], g4r = gt_lds[4][t], g5r = gt_lds[5][t],
              g6r = gt_lds[6][t], g7r = gt_lds[7][t];

  // Left/right neighbors with zero padding at w=0 / w=W-1 (mask instead of halo).
  const int   tm    = (t == 0)      ? 0        : t - 1;
  const int   tp    = (t == kW - 1) ? (kW - 1) : t + 1;
  const float lmask = (t == 0)      ? 0.f : 1.f;
  const float rmask = (t == kW - 1) ? 0.f : 1.f;

  float* outp = out + (size_t)b * kCHW + (size_t)h * kW + t;

  for (int c = 0; c < kC; ++c) {
    const int p = c & 1;
    wait_asynccnt0();    // this wave's async stage of channel c finished
    __syncthreads();     // ...and is visible to every wave
    if (c + 1 < kC) stage_rows(in, in_lds, b, c + 1, h, p ^ 1, t);
    if (c + 2 < kC && t < 192) {     // warm L2 two channels ahead
      const int r = t >> 6, i2 = (t & 63) << 2, hh = h - 1 + r;
      if ((unsigned)hh < (unsigned)kH)
        __builtin_prefetch(in + ((size_t)(b * kC + c + 2) * kH + hh) * kW + i2, 0, 1);
    }

    const float t_m = in_lds[p][0][tm] * lmask;   // (h-1, w-1) TL
    const float t_c = in_lds[p][0][t];            // (h-1, w  ) T
    const float t_p = in_lds[p][0][tp] * rmask;   // (h-1, w+1) TR
    const float m_m = in_lds[p][1][tm] * lmask;   // (h,   w-1) L
    const float m_c = in_lds[p][1][t];            // center
    const float m_p = in_lds[p][1][tp] * rmask;   // (h,   w+1) R
    const float b_m = in_lds[p][2][tm] * lmask;   // (h+1, w-1) BL
    const float b_c = in_lds[p][2][t];            // (h+1, w  ) B
    const float b_p = in_lds[p][2][tp] * rmask;   // (h+1, w+1) BR

    float acc = m_c;                    // same accumulation order as reference
    acc = fmaf(g0r, t_m, acc);
    acc = fmaf(g1r, t_c, acc);
    acc = fmaf(g2r, t_p, acc);
    acc = fmaf(g3r, m_m, acc);
    acc = fmaf(g4r, m_p, acc);
    acc = fmaf(g5r, b_m, acc);
    acc = fmaf(g6r, b_c, acc);
    acc = fmaf(g7r, b_p, acc);

    // Output is write-once: NT store keeps the 128MB input resident in the
    // 192MB L2 so the 3-row stencil reuse never touches HBM twice.
    __builtin_nontemporal_store(acc, &outp[(size_t)c * kHW]);
  }
}

extern "C" void kernel_launch(void* const* d_in, const int* in_sizes, int n_in,
                              void* d_out, int out_size, void* d_ws, size_t ws_size,
                              hipStream_t stream) {
  const float* in = (const float*)d_in[0];   // [8,64,256,256] fp32
  const float* gt = (const float*)d_in[1];   // [8,8,256,256] fp32
  float* out = (float*)d_out;                // [8,64,256,256] fp32
  (void)in_sizes; (void)n_in; (void)out_size; (void)d_ws; (void)ws_size;
  fuse8_kernel<<<dim3(kB * kH), dim3(256), 0, stream>>>(in, gt, out);
}